// VINeuralODE_54769422958765
// MI455X (gfx1250) — compile-verified
//
#include <hip/hip_runtime.h>
#include <math.h>

typedef __attribute__((ext_vector_type(2))) float v2f;
typedef __attribute__((ext_vector_type(8))) float v8f;

// Problem constants
#define Dn 256
#define Hn 1024
#define Bn 8
#define DTc 0.125f

// Workspace layout (float offsets)
static constexpr size_t OFF_G   = 0;                         // H*H  (4 MB)
static constexpr size_t OFF_C   = OFF_G + (size_t)Hn * Hn;   // H
static constexpr size_t OFF_YX  = OFF_C + Hn;                // B*D
static constexpr size_t OFF_YS  = OFF_YX + Bn * Dn;          // B*4  (logq,kl,vfn,jfn)
static constexpr size_t OFF_YIX = OFF_YS + Bn * 4;           // B*D  (stage state)
static constexpr size_t OFF_H   = OFF_YIX + Bn * Dn;         // B*H
static constexpr size_t OFF_S   = OFF_H + (size_t)Bn * Hn;   // B*H
static constexpr size_t OFF_KS  = OFF_S + (size_t)Bn * Hn;   // 6*B*D
static constexpr size_t OFF_ACC = OFF_KS + 6 * Bn * Dn;      // 48 stages * 4 scalars * 8 batch

// Dopri5 tableau
__constant__ float cA[6][5] = {
  {0.f, 0.f, 0.f, 0.f, 0.f},
  {0.2f, 0.f, 0.f, 0.f, 0.f},
  {3.f/40.f, 9.f/40.f, 0.f, 0.f, 0.f},
  {44.f/45.f, -56.f/15.f, 32.f/9.f, 0.f, 0.f},
  {19372.f/6561.f, -25360.f/2187.f, 64448.f/6561.f, -212.f/729.f, 0.f},
  {9017.f/3168.f, -355.f/33.f, 46732.f/5247.f, 49.f/176.f, -5103.f/18656.f},
};
__constant__ float cBW[6] = {35.f/384.f, 0.f, 500.f/1113.f, 125.f/192.f, -2187.f/6784.f, 11.f/84.f};

__device__ inline v8f wmma_f32(v2f a, v2f b, v8f c) {
  // D = A(16x4,f32) * B(4x16,f32) + C(16x16,f32)
  return __builtin_amdgcn_wmma_f32_16x16x4_f32(false, a, false, b, (short)0, c, false, false);
}

// ---------------------------------------------------------------------------
// Init: y_x = x0, y_scal = (log_q0, 0, 0, 0), zero all 48 stage accumulators.
// ---------------------------------------------------------------------------
__global__ void ode_init_kernel(const float* __restrict__ x0, float* __restrict__ ws) {
  int tid = threadIdx.x;
  float* acc = ws + OFF_ACC;
  for (int i = tid; i < 48 * 4 * 8; i += 256) acc[i] = 0.f;
  float* yx = ws + OFF_YX;
  for (int i = tid; i < Bn * Dn; i += 256) yx[i] = x0[i];
  float* ysc = ws + OFF_YS;
  if (tid < 32 && (tid & 3) != 0) ysc[tid] = 0.f;
  __shared__ float red[256];
  for (int b = 0; b < Bn; ++b) {
    float v = x0[b * Dn + tid];
    red[tid] = v * v;
    __syncthreads();
    for (int off = 128; off > 0; off >>= 1) {
      if (tid < off) red[tid] += red[tid + off];
      __syncthreads();
    }
    if (tid == 0) ysc[b * 4] = -0.5f * (256.f * 1.8378770664093453f + red[0]); // D*ln(2pi)
    __syncthreads();
  }
}

// ---------------------------------------------------------------------------
// c[k] = sum_i W1[i,k] * W2[k,i]   (trace reduction vector)
// ---------------------------------------------------------------------------
__global__ void ode_cvec_kernel(const float* __restrict__ W1, const float* __restrict__ W2,
                                float* __restrict__ ws) {
  int k = blockIdx.x * 256 + threadIdx.x;
  float a = 0.f;
  for (int i = 0; i < Dn; ++i) a = fmaf(W1[i * Hn + k], W2[k * Dn + i], a);
  ws[OFF_C + k] = a;
}

// ---------------------------------------------------------------------------
// G = (W1^T W1) .* (W2 W2^T), 1024x1024, via v_wmma_f32_16x16x4_f32.
// One wave per 16x16 tile; two independent WMMA accumulators per K-step.
// ---------------------------------------------------------------------------
__global__ void ode_gram_kernel(const float* __restrict__ W1, const float* __restrict__ W2,
                                float* __restrict__ ws) {
  float* G = ws + OFF_G;
  int lane = threadIdx.x & 31;
  int wave = threadIdx.x >> 5;
  int tile = blockIdx.x * 8 + wave;          // 4096 tiles total
  int tp = (tile >> 6) << 4;
  int tq = (tile & 63) << 4;
  int m = lane & 15, kh = lane >> 4;
  v8f g1 = {};
  v8f g2 = {};
  for (int k0 = 0; k0 < Dn; k0 += 4) {
    int ka = k0 + 2 * kh;
    v2f a1, b1, a2, b2;
    // Gram1 = W1^T W1: A[m][kk]=W1[kk, tp+m], B[kk][n]=W1[kk, tq+n]
    a1.x = W1[(ka    ) * Hn + tp + m];
    a1.y = W1[(ka + 1) * Hn + tp + m];
    b1.x = W1[(ka    ) * Hn + tq + m];
    b1.y = W1[(ka + 1) * Hn + tq + m];
    // Gram2 = W2 W2^T: A[m][kk]=W2[tp+m, kk], B[kk][n]=W2[tq+n, kk]
    a2.x = W2[(tp + m) * Dn + ka];
    a2.y = W2[(tp + m) * Dn + ka + 1];
    b2.x = W2[(tq + m) * Dn + ka];
    b2.y = W2[(tq + m) * Dn + ka + 1];
    g1 = wmma_f32(a1, b1, g1);
    g2 = wmma_f32(a2, b2, g2);
  }
  int col = tq + m;
#pragma unroll
  for (int r = 0; r < 8; ++r) {
    int row = tp + r + 8 * kh;
    G[row * Hn + col] = g1[r] * g2[r];
  }
}

// ---------------------------------------------------------------------------
// Stage part 1: yi = y + DT*sum(a_ij k_j); u = yi@W1 + b1 + wt*t;
// h=tanh(u), s=1-h^2; accumulate dlogq = -sum(s*c).
// grid (B=8, 8 column chunks) x 128 threads.
// ---------------------------------------------------------------------------
__global__ void ode_stage_act_kernel(const float* __restrict__ W1, const float* __restrict__ b1,
                                     const float* __restrict__ wt, float* __restrict__ ws,
                                     float t, int stage, int g) {
  int b = blockIdx.x;
  int chunk = blockIdx.y;
  int tid = threadIdx.x;
  const float* yx = ws + OFF_YX;
  const float* ks = ws + OFF_KS;
  __shared__ float yiS[Dn];
  __shared__ float red[128];
  for (int d = tid; d < Dn; d += 128) {
    float v = yx[b * Dn + d];
    for (int j = 0; j < stage; ++j)
      v = fmaf(DTc * cA[stage][j], ks[j * (Bn * Dn) + b * Dn + d], v);
    yiS[d] = v;
    if (chunk == 0) ws[OFF_YIX + b * Dn + d] = v;   // stage state for dkl
  }
  __syncthreads();
  int col = chunk * 128 + tid;
  float u = fmaf(wt[col], t, b1[col]);
  for (int d = 0; d < Dn; ++d) u = fmaf(yiS[d], W1[d * Hn + col], u);
  float hh = tanhf(u);
  float ss = 1.f - hh * hh;
  ws[OFF_H + b * Hn + col] = hh;
  ws[OFF_S + b * Hn + col] = ss;
  red[tid] = ss * ws[OFF_C + col];
  __syncthreads();
  for (int off = 64; off > 0; off >>= 1) {
    if (tid < off) red[tid] += red[tid + off];
    __syncthreads();
  }
  if (tid == 0) atomicAdd(ws + OFF_ACC + (size_t)g * 32 + 0 * 8 + b, -red[0]);
}

// ---------------------------------------------------------------------------
// Stage part 2 (WMMA): waves 0..63 compute Gs = S_pad(16xH) @ G tile-by-tile
// and reduce djfn = s.Gs; waves 64..79 compute dxdt = H_pad @ W2 + b2, store
// k_i, and reduce dvfn = |dxdt|^2, dkl = <yi - mu, dxdt>.
// grid 10 x 256 (80 waves), batch rows padded 8 -> 16 with zeros.
// ---------------------------------------------------------------------------
__global__ void ode_stage_heavy_kernel(const float* __restrict__ W2, const float* __restrict__ b2,
                                       const float* __restrict__ mu, float* __restrict__ ws,
                                       int stage, int g) {
  int lane = threadIdx.x & 31;
  int wave = (blockIdx.x << 3) + (threadIdx.x >> 5);   // 0..79, wave-uniform
  int n = lane & 15, kh = lane >> 4;
  const float* S = ws + OFF_S;
  const float* Hh = ws + OFF_H;
  const float* G = ws + OFF_G;
  float* acc = ws + OFF_ACC + (size_t)g * 32;

  if (wave < 64) {
    int col = (wave << 4) + n;
    v8f d = {};
    for (int k0 = 0; k0 < Hn; k0 += 4) {
      int ka = k0 + 2 * kh;
      v2f a, b;
      a.x = (n < 8) ? S[n * Hn + ka    ] : 0.f;
      a.y = (n < 8) ? S[n * Hn + ka + 1] : 0.f;
      b.x = G[(ka    ) * Hn + col];
      b.y = G[(ka + 1) * Hn + col];
      d = wmma_f32(a, b, d);
    }
#pragma unroll
    for (int r = 0; r < 8; ++r) {                     // r == batch row
      float p = (lane < 16) ? S[r * Hn + col] * d[r] : 0.f;
      p += __shfl_xor(p, 8, 16);
      p += __shfl_xor(p, 4, 16);
      p += __shfl_xor(p, 2, 16);
      p += __shfl_xor(p, 1, 16);
      if (lane == 0) atomicAdd(acc + 3 * 8 + r, p);   // djfn
    }
  } else {
    int col = ((wave - 64) << 4) + n;                 // 0..255
    v8f d = {};
    for (int k0 = 0; k0 < Hn; k0 += 4) {
      int ka = k0 + 2 * kh;
      v2f a, b;
      a.x = (n < 8) ? Hh[n * Hn + ka    ] : 0.f;
      a.y = (n < 8) ? Hh[n * Hn + ka + 1] : 0.f;
      b.x = W2[(ka    ) * Dn + col];
      b.y = W2[(ka + 1) * Dn + col];
      d = wmma_f32(a, b, d);
    }
    float* ksx = ws + OFF_KS + (size_t)stage * (Bn * Dn);
    const float* yix = ws + OFF_YIX;
#pragma unroll
    for (int r = 0; r < 8; ++r) {
      float dx = d[r] + b2[col];
      float pv = 0.f, pk = 0.f;
      if (lane < 16) {
        ksx[r * Dn + col] = dx;
        pv = dx * dx;
        pk = (yix[r * Dn + col] - mu[col]) * dx;
      }
      pv += __shfl_xor(pv, 8, 16); pv += __shfl_xor(pv, 4, 16);
      pv += __shfl_xor(pv, 2, 16); pv += __shfl_xor(pv, 1, 16);
      pk += __shfl_xor(pk, 8, 16); pk += __shfl_xor(pk, 4, 16);
      pk += __shfl_xor(pk, 2, 16); pk += __shfl_xor(pk, 1, 16);
      if (lane == 0) {
        atomicAdd(acc + 2 * 8 + r, pv);               // dvfn
        atomicAdd(acc + 1 * 8 + r, pk);               // dkl
      }
    }
  }
}

// ---------------------------------------------------------------------------
// Combine: y += DT * sum_i BW[i] * k_i (x-part and scalar part).
// ---------------------------------------------------------------------------
__global__ void ode_combine_kernel(float* __restrict__ ws, int step) {
  int idx = blockIdx.x * 256 + threadIdx.x;   // 0..2047
  float* yx = ws + OFF_YX;
  const float* ks = ws + OFF_KS;
  float v = yx[idx];
#pragma unroll
  for (int i = 0; i < 6; ++i) v = fmaf(DTc * cBW[i], ks[i * (Bn * Dn) + idx], v);
  yx[idx] = v;
  if (blockIdx.x == 0 && threadIdx.x < 32) {
    int b = threadIdx.x >> 2, sc = threadIdx.x & 3;
    float* ysc = ws + OFF_YS;
    const float* acc = ws + OFF_ACC;
    float v2 = ysc[threadIdx.x];
#pragma unroll
    for (int i = 0; i < 6; ++i)
      v2 = fmaf(DTc * cBW[i], acc[(size_t)(step * 6 + i) * 32 + sc * 8 + b], v2);
    ysc[threadIdx.x] = v2;
  }
}

// ---------------------------------------------------------------------------
// Finalize: out[b, 0..255] = z, out[b, 256..259] = (logq, kl, vfn, jfn)
// ---------------------------------------------------------------------------
__global__ void ode_finalize_kernel(const float* __restrict__ ws, float* __restrict__ out) {
  for (int idx = threadIdx.x; idx < Bn * (Dn + 4); idx += 256) {
    int b = idx / (Dn + 4), j = idx % (Dn + 4);
    out[idx] = (j < Dn) ? ws[OFF_YX + b * Dn + j] : ws[OFF_YS + b * 4 + (j - Dn)];
  }
}

extern "C" void kernel_launch(void* const* d_in, const int* in_sizes, int n_in,
                              void* d_out, int out_size, void* d_ws, size_t ws_size,
                              hipStream_t stream) {
  (void)in_sizes; (void)n_in; (void)out_size; (void)ws_size;
  const float* x0 = (const float*)d_in[0];
  const float* W1 = (const float*)d_in[1];
  const float* b1 = (const float*)d_in[2];
  const float* wt = (const float*)d_in[3];
  const float* W2 = (const float*)d_in[4];
  const float* b2 = (const float*)d_in[5];
  const float* mu = (const float*)d_in[6];
  float* ws = (float*)d_ws;
  float* out = (float*)d_out;
  static const float Ch[6] = {0.f, 0.2f, 0.3f, 0.8f, 8.f / 9.f, 1.f};

  ode_init_kernel<<<1, 256, 0, stream>>>(x0, ws);
  ode_cvec_kernel<<<4, 256, 0, stream>>>(W1, W2, ws);
  ode_gram_kernel<<<512, 256, 0, stream>>>(W1, W2, ws);
  for (int n_step = 0; n_step < 8; ++n_step) {
    for (int i = 0; i < 6; ++i) {
      float tt = ((float)n_step + Ch[i]) * DTc;
      int g = n_step * 6 + i;
      ode_stage_act_kernel<<<dim3(8, 8), 128, 0, stream>>>(W1, b1, wt, ws, tt, i, g);
      ode_stage_heavy_kernel<<<10, 256, 0, stream>>>(W2, b2, mu, ws, i, g);
    }
    ode_combine_kernel<<<8, 256, 0, stream>>>(ws, n_step);
  }
  ode_finalize_kernel<<<1, 256, 0, stream>>>(ws, out);
}